// RKNLayer_18313740550768
// MI455X (gfx1250) — compile-verified
//
#include <hip/hip_runtime.h>
#include <hip/hip_bf16.h>

// RKN banded Kalman scan for MI455X (gfx1250).
//
// Dims: B=128 batches, T=128 steps (serial), LOD=128, LSD=256, K=16 mixture,
// band half-width 3 -> 7 diagonals per transition matrix.
//
// Strategy:
//  * Kernel 1: extract the 7 nonzero diagonals of tm11/12/21/22 into a compact
//    TMB[n][k] (n = (mat*128+i)*7+d, k-contiguous) in d_ws. This is the WMMA
//    B-matrix for the per-step mixture GEMM.
//  * Kernel 2: 8 workgroups x 256 threads (8 wave32). Each WG owns 16 batches
//    and runs the full T=128 scan with state in LDS (ping-pong buffers).
//    Per step:
//      ph1: elementwise Kalman posterior update (+ emit outputs)
//      ph2: logits = post_mean @ coeff_w via v_wmma_f32_16x16x4_f32 chains,
//           partials merged with ds_add_f32; softmax -> alpha
//           (coeff_w B-fragments hoisted to registers outside the t-loop)
//      ph3: Aband(16x3584) = alpha(16x16) x TMB(16x3584) via WMMA f32 tiles,
//           unroll-4 so TMB b64 loads from L2 overlap the WMMA chains
//      ph4: banded matvecs for next prior mean/covariances (branch-free,
//           identity folded in at the main diagonal d==3) + prefetch of the
//           next timestep's observation rows (global_prefetch_b8)

#define LOD 128
#define LSD 256
#define NK  16
#define NB  128
#define NT  128
#define BW  3
#define ND  7              // diagonals
#define BPW 16             // batches per workgroup
#define NWG (NB / BPW)     // 8 workgroups
#define NBAND (4 * LOD * ND)   // 3584 columns of the mixture GEMM
#define TILES (NBAND / 16)     // 224 WMMA N-tiles
#define NWAVES 8
#define TPW (TILES / NWAVES)   // 28 tiles per wave

typedef float v2f __attribute__((ext_vector_type(2)));
typedef float v8f __attribute__((ext_vector_type(8)));

// ---------------- Kernel 1: band extraction -> TMB[n][k] ----------------
__global__ void rkn_extract_bands(const float* __restrict__ tm11,
                                  const float* __restrict__ tm12,
                                  const float* __restrict__ tm21,
                                  const float* __restrict__ tm22,
                                  float* __restrict__ tmb) {
    int idx = blockIdx.x * blockDim.x + threadIdx.x;     // over NBAND*NK
    if (idx >= NBAND * NK) return;
    int k = idx & (NK - 1);
    int n = idx >> 4;
    int d = n % ND;
    int r = n / ND;
    int i = r & (LOD - 1);
    int mat = r >> 7;
    int c = i - BW + d;
    float v = 0.0f;
    if (c >= 0 && c < LOD) {
        const float* src = (mat == 0) ? tm11 : (mat == 1) ? tm12
                         : (mat == 2) ? tm21 : tm22;
        v = src[(k * LOD + i) * LOD + c];
    }
    tmb[n * NK + k] = v;   // k-contiguous: one b64 per lane loads a K-pair
}

// ---------------- Kernel 2: the scan ----------------
__global__ __launch_bounds__(256, 1)
void rkn_scan_kernel(const float* __restrict__ latent_obs,
                     const float* __restrict__ obs_vars,
                     const float* __restrict__ initial_mean,
                     const float* __restrict__ icu,
                     const float* __restrict__ icl,
                     const float* __restrict__ ics,
                     const float* __restrict__ coeff_w,   // [LSD][NK]
                     const float* __restrict__ coeff_b,   // [NK]
                     const float* __restrict__ tcu,       // [LOD]
                     const float* __restrict__ tcl,       // [LOD]
                     const float* __restrict__ tmb,       // [NBAND][NK]
                     float* __restrict__ out) {
    extern __shared__ float smem[];
    // dynamic LDS carve-up (floats); total ~313 KB of the 320 KB WGP LDS
    float* S_pm     = smem;                 // [2][BPW][LSD]   8192
    float* S_cu     = S_pm + 2 * BPW * LSD; // [2][BPW][LOD]   4096
    float* S_cl     = S_cu + 2 * BPW * LOD;
    float* S_cs     = S_cl + 2 * BPW * LOD;
    float* S_logits = S_cs + 2 * BPW * LOD; // [BPW][NK]        256
    float* S_alpha  = S_logits + BPW * NK;  // [BPW][NK]        256
    float* S_aband  = S_alpha + BPW * NK;   // [BPW][NBAND]   57344

    const int tid    = threadIdx.x;
    const int lane   = tid & 31;
    const int wave   = tid >> 5;
    const int laneLo = lane & 15;
    const int laneHi = lane >> 4;           // 0 or 1
    const int wg     = blockIdx.x;          // 0..7

    const int b  = tid >> 4;                // local batch 0..15
    const int i0 = (tid & 15) * 8;          // 8 consecutive rows per thread
    const int bg = wg * BPW + b;            // global batch

    // output section bases (floats)
    const size_t OPM = 0;
    const size_t OCU = (size_t)NB * NT * LSD;
    const size_t OCL = OCU + (size_t)NB * NT * LOD;
    const size_t OCS = OCL + (size_t)NB * NT * LOD;

    // hoist per-thread transition-cov constants
    float tcu_r[8], tcl_r[8];
#pragma unroll
    for (int ii = 0; ii < 8; ++ii) { tcu_r[ii] = tcu[i0 + ii]; tcl_r[ii] = tcl[i0 + ii]; }

    // hoist the logits-GEMM B fragments (coeff_w is t-invariant): 16 VGPRs/lane
    v2f cwf[8];
    {
        int jbase = wave * 32;
#pragma unroll
        for (int kc = 0; kc < 8; ++kc) {
            int jr = jbase + kc * 4 + laneHi * 2;
            v2f bf;
            bf.x = coeff_w[jr * NK + laneLo];
            bf.y = coeff_w[(jr + 1) * NK + laneLo];
            cwf[kc] = bf;
        }
    }

    // ---- load initial state into parity-0 buffers ----
#pragma unroll
    for (int ii = 0; ii < 8; ++ii) {
        int i = i0 + ii;
        S_pm[b * LSD + i]        = initial_mean[bg * LSD + i];
        S_pm[b * LSD + LOD + i]  = initial_mean[bg * LSD + LOD + i];
        S_cu[b * LOD + i]        = icu[bg * LOD + i];
        S_cl[b * LOD + i]        = icl[bg * LOD + i];
        S_cs[b * LOD + i]        = ics[bg * LOD + i];
    }
    __syncthreads();

    for (int t = 0; t < NT; ++t) {
        const int p = t & 1;
        float* pmP = S_pm + p * BPW * LSD;
        float* cuP = S_cu + p * BPW * LOD;
        float* clP = S_cl + p * BPW * LOD;
        float* csP = S_cs + p * BPW * LOD;
        float* pmN = S_pm + (p ^ 1) * BPW * LSD;
        float* cuN = S_cu + (p ^ 1) * BPW * LOD;
        float* clN = S_cl + (p ^ 1) * BPW * LOD;
        float* csN = S_cs + (p ^ 1) * BPW * LOD;

        // ---------- phase 1: posterior update (in-place) + outputs ----------
        S_logits[tid & 255] = coeff_b[tid & 15];   // bias-init logits buffer
        {
            const size_t obase = ((size_t)bg * NT + t) * LOD;
#pragma unroll
            for (int ii = 0; ii < 8; ++ii) {
                int i = i0 + ii;
                float obs = latent_obs[obase + i];
                float ov  = obs_vars[obase + i];
                float pmu = pmP[b * LSD + i];
                float pml = pmP[b * LSD + LOD + i];
                float cuv = cuP[b * LOD + i];
                float clv = clP[b * LOD + i];
                float csv = csP[b * LOD + i];
                float denom = cuv + ov;
                float inv   = 1.0f / denom;
                float qu = cuv * inv;
                float ql = csv * inv;
                float res = obs - pmu;
                float postu = pmu + qu * res;
                float postl = pml + ql * res;
                float cf  = 1.0f - qu;
                float pcu = cf * cuv;
                float pcl = clv - ql * csv;
                float pcs = cf * csv;
                pmP[b * LSD + i]       = postu;
                pmP[b * LSD + LOD + i] = postl;
                cuP[b * LOD + i] = pcu;
                clP[b * LOD + i] = pcl;
                csP[b * LOD + i] = pcs;
                size_t o2 = ((size_t)bg * NT + t);
                out[OPM + o2 * LSD + i]        = postu;
                out[OPM + o2 * LSD + LOD + i]  = postl;
                out[OCU + o2 * LOD + i] = pcu;
                out[OCL + o2 * LOD + i] = pcl;
                out[OCS + o2 * LOD + i] = pcs;
            }
        }
        __syncthreads();

        // ---------- phase 2: logits = post_mean @ coeff_w (WMMA) ----------
        {
            v8f c = {};
            int jbase = wave * 32;             // this wave's K-slice of 32
#pragma unroll
            for (int kc = 0; kc < 8; ++kc) {
                int j0 = jbase + kc * 4;
                // A frag: post_mean rows (M=batch), K-pair per half-wave
                v2f a = *(const v2f*)&pmP[laneLo * LSD + j0 + laneHi * 2];
                c = __builtin_amdgcn_wmma_f32_16x16x4_f32(
                        false, a, false, cwf[kc], (short)0, c, false, false);
            }
#pragma unroll
            for (int j = 0; j < 8; ++j) {
                int m = j + laneHi * 8;
                atomicAdd(&S_logits[m * NK + laneLo], c[j]);   // ds_add_f32
            }
        }
        __syncthreads();

        // ---------- softmax -> alpha ----------
        if (tid < BPW) {
            float mx = -3.0e38f;
#pragma unroll
            for (int k = 0; k < NK; ++k) mx = fmaxf(mx, S_logits[tid * NK + k]);
            float s = 0.0f;
            float e[NK];
#pragma unroll
            for (int k = 0; k < NK; ++k) { e[k] = __expf(S_logits[tid * NK + k] - mx); s += e[k]; }
            float invs = 1.0f / s;
#pragma unroll
            for (int k = 0; k < NK; ++k) S_alpha[tid * NK + k] = e[k] * invs;
        }
        __syncthreads();

        // ---------- phase 3: Aband = alpha x TMB (WMMA GEMM, M=16,K=16,N=3584) ----------
        {
            // alpha fragments are tiny and t-varying; load once per step
            v2f af[4];
#pragma unroll
            for (int kc = 0; kc < 4; ++kc)
                af[kc] = *(const v2f*)&S_alpha[laneLo * NK + kc * 4 + laneHi * 2];

            // unroll 4: clause up 16 global b64 loads per iteration group and
            // overlap them with the previous tiles' WMMA chains
#pragma unroll 4
            for (int tt = 0; tt < TPW; ++tt) {
                int n0 = (wave * TPW + tt) * 16;
                v8f c = {};
#pragma unroll
                for (int kc = 0; kc < 4; ++kc) {
                    v2f bf = *(const v2f*)&tmb[(size_t)(n0 + laneLo) * NK + kc * 4 + laneHi * 2];
                    c = __builtin_amdgcn_wmma_f32_16x16x4_f32(
                            false, af[kc], false, bf, (short)0, c, false, false);
                }
#pragma unroll
                for (int j = 0; j < 8; ++j) {
                    int m = j + laneHi * 8;                  // batch row
                    S_aband[m * NBAND + n0 + laneLo] = c[j];
                }
            }
        }
        __syncthreads();

        // ---------- phase 4: banded matvecs -> next prior ----------
        {
            // prefetch next timestep's observation rows while we crunch VALU
            if (t + 1 < NT) {
                const size_t onext = ((size_t)bg * NT + (t + 1)) * LOD + i0;
                __builtin_prefetch(&latent_obs[onext], 0, 3);  // global_prefetch_b8
                __builtin_prefetch(&obs_vars[onext], 0, 3);
            }

            // cache sliding windows c = i0-3 .. i0+10 (14 values per array)
            float wu[14], wl[14], wpu[14], wpl[14], wps[14];
#pragma unroll
            for (int cc = 0; cc < 14; ++cc) {
                int c = i0 - BW + cc;
                bool ok = (c >= 0) && (c < LOD);
                wu[cc]  = ok ? pmP[b * LSD + c]       : 0.0f;
                wl[cc]  = ok ? pmP[b * LSD + LOD + c] : 0.0f;
                wpu[cc] = ok ? cuP[b * LOD + c] : 0.0f;
                wpl[cc] = ok ? clP[b * LOD + c] : 0.0f;
                wps[cc] = ok ? csP[b * LOD + c] : 0.0f;
            }
            const float* Ab = S_aband + b * NBAND;
#pragma unroll
            for (int ii = 0; ii < 8; ++ii) {
                int i = i0 + ii;
                float nu = 0.f, nl = 0.f, au = 0.f, al = 0.f, as_ = 0.f;
#pragma unroll
                for (int d = 0; d < ND; ++d) {
                    int base = i * ND + d;
                    float a11 = Ab[base];                     // mat0
                    float a12 = Ab[896 + base];               // mat1
                    float a21 = Ab[2 * 896 + base];           // mat2
                    float a22 = Ab[3 * 896 + base];           // mat3
                    float eye = (d == BW) ? 1.0f : 0.0f;      // fold identity
                    float a11e = a11 + eye;
                    float a22e = a22 + eye;
                    int w = ii + d;
                    float mu = wu[w], ml = wl[w];
                    float pu = wpu[w], pl = wpl[w], ps = wps[w];
                    nu += a11e * mu + a12 * ml;
                    nl += a21 * mu + a22e * ml;
                    au += a11e * a11e * pu + 2.0f * a11e * a12 * ps + a12 * a12 * pl;
                    al += a21 * a21 * pu + 2.0f * a21 * a22e * ps + a22e * a22e * pl;
                    as_ += a21 * a11e * pu + (a22e * a11e + a21 * a12) * ps + a22e * a12 * pl;
                }
                pmN[b * LSD + i]       = nu;
                pmN[b * LSD + LOD + i] = nl;
                cuN[b * LOD + i] = au + tcu_r[ii];
                clN[b * LOD + i] = al + tcl_r[ii];
                csN[b * LOD + i] = as_;
            }
        }
        __syncthreads();
    }
}

extern "C" void kernel_launch(void* const* d_in, const int* in_sizes, int n_in,
                              void* d_out, int out_size, void* d_ws, size_t ws_size,
                              hipStream_t stream) {
    const float* latent_obs = (const float*)d_in[0];
    const float* obs_vars   = (const float*)d_in[1];
    const float* init_mean  = (const float*)d_in[2];
    const float* icu        = (const float*)d_in[3];
    const float* icl        = (const float*)d_in[4];
    const float* ics        = (const float*)d_in[5];
    const float* tm11       = (const float*)d_in[6];
    const float* tm12       = (const float*)d_in[7];
    const float* tm21       = (const float*)d_in[8];
    const float* tm22       = (const float*)d_in[9];
    const float* coeff_w    = (const float*)d_in[10];
    const float* coeff_b    = (const float*)d_in[11];
    const float* tcu        = (const float*)d_in[12];
    const float* tcl        = (const float*)d_in[13];
    float* out = (float*)d_out;
    float* tmb = (float*)d_ws;        // NBAND*NK floats = 224 KB

    // 1) extract banded transition matrices into WMMA-friendly layout
    {
        int total = NBAND * NK;
        int blocks = (total + 255) / 256;
        rkn_extract_bands<<<blocks, 256, 0, stream>>>(tm11, tm12, tm21, tm22, tmb);
    }
    // 2) scan: 8 WGs x 256 threads, ~313 KB dynamic LDS each (one WG / WGP)
    {
        size_t smem_bytes = (size_t)(2 * BPW * LSD + 3 * 2 * BPW * LOD +
                                     2 * BPW * NK + BPW * NBAND) * sizeof(float);
        rkn_scan_kernel<<<NWG, 256, smem_bytes, stream>>>(
            latent_obs, obs_vars, init_mean, icu, icl, ics,
            coeff_w, coeff_b, tcu, tcl, tmb, out);
    }
}